// STGCN_90855738179679
// MI455X (gfx1250) — compile-verified
//
#include <hip/hip_runtime.h>
#include <cstddef>
#include <cstdint>

// ---------------------------------------------------------------------------
// STGCN forward for MI455X (gfx1250, wave32). All GEMMs run on the fp32 WMMA
// V_WMMA_F32_16X16X4_F32. LDS rows padded to 68 floats -> every fragment is a
// single aligned ds_load_b64 (compiler fuses pairs into ds_load_2addr_b64).
// Graph GEMM: BM=128/BK=32, double-buffered LDS, 1 barrier per stage,
// global loads overlapped with WMMA. launch_bounds(128,1) prevents spills.
// ---------------------------------------------------------------------------

typedef __attribute__((ext_vector_type(2))) float v2f;
typedef __attribute__((ext_vector_type(8))) float v8f;

#define N_NODES 2048
#define HID     64
#define BATCH   16
#define PAD     68   // LDS row stride (floats): 16B-multiple, conflict-free

static __device__ __forceinline__ v8f wmma_f32x4(v2f a, v2f b, v8f c) {
  return __builtin_amdgcn_wmma_f32_16x16x4_f32(false, a, false, b, (short)0, c,
                                               false, false);
}

static __device__ __forceinline__ v8f vzero8() {
  v8f z;
#pragma unroll
  for (int r = 0; r < 8; ++r) z[r] = 0.f;
  return z;
}

// ---------------- Laplacian normalization ----------------------------------
__global__ void k_dis(const float* __restrict__ adj, float* __restrict__ dis) {
  __shared__ float red[256];
  const int i = blockIdx.x;
  const int tid = threadIdx.x;
  float s = 0.f;
  for (int j = tid; j < N_NODES; j += 256)
    if (j != i) s += adj[(size_t)i * N_NODES + j];
  red[tid] = s;
  __syncthreads();
  for (int off = 128; off > 0; off >>= 1) {
    if (tid < off) red[tid] += red[tid + off];
    __syncthreads();
  }
  if (tid == 0) {
    float deg = red[0];
    dis[i] = (deg > 0.f) ? rsqrtf(fmaxf(deg, 1e-30f)) : 0.f;
  }
}

__global__ void k_norm(const float* __restrict__ adj,
                       const float* __restrict__ dis,
                       float* __restrict__ nrm) {
  size_t idx = (size_t)blockIdx.x * 256 + threadIdx.x;
  if (idx >= (size_t)N_NODES * N_NODES) return;
  int i = (int)(idx >> 11);
  int j = (int)(idx & 2047);
  float a = (i == j) ? 0.f : adj[idx];
  nrm[idx] = -(dis[i] * a * dis[j]);
}

// ---------------- input permute (B,2,N,24) -> (B,24,N,2) --------------------
__global__ void k_permute(const float* __restrict__ x, float* __restrict__ x0) {
  int idx = blockIdx.x * 256 + threadIdx.x;
  const int total = BATCH * 24 * N_NODES * 2;
  if (idx >= total) return;
  int c = idx & 1;
  int n = (idx >> 1) & 2047;
  int t = (idx >> 12) % 24;
  int b = idx / (24 * N_NODES * 2);
  x0[idx] = x[(((size_t)b * 2 + c) * N_NODES + n) * 24 + t];
}

// ---------------- transpose a batch of 64x64 matrices -----------------------
__global__ void k_transpose64(const float* __restrict__ src,
                              float* __restrict__ dst) {
  const float* s = src + (size_t)blockIdx.x * 4096;
  float* d = dst + (size_t)blockIdx.x * 4096;
  for (int idx = threadIdx.x; idx < 4096; idx += 256) {
    int r = idx >> 6, c = idx & 63;
    d[c * 64 + r] = s[idx];
  }
}

// ---------------- temporal GLU conv, cin=2 (tiny, VALU) ---------------------
__global__ void k_tc_cin2(const float* __restrict__ in,
                          const float* __restrict__ w1, const float* __restrict__ b1,
                          const float* __restrict__ w2, const float* __restrict__ b2,
                          const float* __restrict__ w3, const float* __restrict__ b3,
                          float* __restrict__ out) {
  int i = blockIdx.x * 256 + threadIdx.x;
  const int total = BATCH * 22 * N_NODES * 64;
  if (i >= total) return;
  int co = i & 63;
  int r = i >> 6;
  int n = r & 2047;
  int q = r >> 11;
  int so = q % 22;
  int b  = q / 22;
  float c1 = b1[co], c2 = b2[co], c3 = b3[co];
#pragma unroll
  for (int dt = 0; dt < 3; ++dt) {
#pragma unroll
    for (int ci = 0; ci < 2; ++ci) {
      float xv = in[(((size_t)b * 24 + so + dt) * N_NODES + n) * 2 + ci];
      int wi = (dt * 2 + ci) * 64 + co;
      c1 = fmaf(xv, w1[wi], c1);
      c2 = fmaf(xv, w2[wi], c2);
      c3 = fmaf(xv, w3[wi], c3);
    }
  }
  float g = c1 * (1.f / (1.f + __expf(-c2))) + c3;
  out[i] = fmaxf(g, 0.f);
}

// ---------------- batched graph GEMM:  Y[g] = alpha*norm@X[g] + beta*Z[g] ---
// norm: 2048x2048, X[g]/Y[g]/Z[g]: 2048x64 row-major. BM=128, BK=32.
// Double-buffered LDS: one barrier/stage; next-stage global loads issue
// before the WMMA burst so HBM latency hides under the matrix pipe.
__global__ __launch_bounds__(128, 1)
void k_graph_gemm(const float* __restrict__ nrm, const float* __restrict__ X,
                  const float* __restrict__ Z, float* __restrict__ Y,
                  float alpha, float beta) {
  const int m0 = blockIdx.x * 128;
  const int g  = blockIdx.y;
  const float* Xg = X + (size_t)g * N_NODES * HID;

  __shared__ float As[2][128][PAD];   // [buf][m][k]
  __shared__ float Bt[2][64][PAD];    // [buf][col][k] (transposed B)

  const int tid = threadIdx.x;
  const int wave = tid >> 5, lane = tid & 31;
  const int mrow = lane & 15;
  const int khalf = (lane >> 4) * 2;

  v8f acc0[4], acc1[4];
#pragma unroll
  for (int t = 0; t < 4; ++t) { acc0[t] = vzero8(); acc1[t] = vzero8(); }

  float4 ra[8];
  float rbx[8], rby[8];

  // prologue: load stage 0 into registers
#pragma unroll
  for (int rep = 0; rep < 8; ++rep) {
    int idx = tid + rep * 128;                 // 0..1023
    int r = idx >> 3, c = (idx & 7) * 4;       // A: 128 rows x 32 k
    ra[rep] = *reinterpret_cast<const float4*>(
        nrm + (size_t)(m0 + r) * N_NODES + c);
    int k2 = idx >> 6, cc = idx & 63;          // B pairs: 16 k-pairs x 64 cols
    const float* bp = Xg + (size_t)(2 * k2) * HID + cc;
    rbx[rep] = bp[0];
    rby[rep] = bp[HID];
  }

  int p = 0;
  for (int k0 = 0; k0 < N_NODES; k0 += 32, p ^= 1) {
    // regs -> LDS buffer p
#pragma unroll
    for (int rep = 0; rep < 8; ++rep) {
      int idx = tid + rep * 128;
      int r = idx >> 3, c = (idx & 7) * 4;
      *reinterpret_cast<float4*>(&As[p][r][c]) = ra[rep];
      int k2 = idx >> 6, cc = idx & 63;
      v2f pr; pr.x = rbx[rep]; pr.y = rby[rep];
      *reinterpret_cast<v2f*>(&Bt[p][cc][2 * k2]) = pr;
    }
    __syncthreads();
    // issue next-stage global loads (consumed next iteration)
    if (k0 + 32 < N_NODES) {
#pragma unroll
      for (int rep = 0; rep < 8; ++rep) {
        int idx = tid + rep * 128;
        int r = idx >> 3, c = (idx & 7) * 4;
        ra[rep] = *reinterpret_cast<const float4*>(
            nrm + (size_t)(m0 + r) * N_NODES + k0 + 32 + c);
        int k2 = idx >> 6, cc = idx & 63;
        const float* bp = Xg + (size_t)(k0 + 32 + 2 * k2) * HID + cc;
        rbx[rep] = bp[0];
        rby[rep] = bp[HID];
      }
    }
    // compute on LDS buffer p
#pragma unroll
    for (int kk = 0; kk < 32; kk += 4) {
      v2f a0 = *reinterpret_cast<const v2f*>(&As[p][wave * 32 + mrow][kk + khalf]);
      v2f a1 = *reinterpret_cast<const v2f*>(&As[p][wave * 32 + 16 + mrow][kk + khalf]);
#pragma unroll
      for (int t = 0; t < 4; ++t) {
        v2f b = *reinterpret_cast<const v2f*>(&Bt[p][t * 16 + mrow][kk + khalf]);
        acc0[t] = wmma_f32x4(a0, b, acc0[t]);
        acc1[t] = wmma_f32x4(a1, b, acc1[t]);
      }
    }
  }

  // epilogue. C/D layout: vgpr r -> row r + 8*(lane>=16), col = lane&15
  const int n = lane & 15;
  const size_t gbase = (size_t)g * N_NODES * HID;
  const int mb0 = m0 + wave * 32 + (lane >> 4) * 8;
#pragma unroll
  for (int t = 0; t < 4; ++t) {
#pragma unroll
    for (int r = 0; r < 8; ++r) {
      size_t off = gbase + (size_t)(mb0 + r) * HID + t * 16 + n;
      float v = alpha * acc0[t][r];
      if (beta != 0.f) v = fmaf(beta, Z[off], v);
      Y[off] = v;
      size_t off2 = off + (size_t)16 * HID;   // second 16-row subtile
      float v2 = alpha * acc1[t][r];
      if (beta != 0.f) v2 = fmaf(beta, Z[off2], v2);
      Y[off2] = v2;
    }
  }
}

// ---------------- channel GEMM:  Out = [Out +] X@Wt^T [+ bias] [relu] -------
// X: M x 64 row-major (M multiple of 128), Wt: PRE-TRANSPOSED 64x64 ([f][k]).
__global__ __launch_bounds__(128, 1)
void k_chan_gemm(const float* __restrict__ X, const float* __restrict__ Wt,
                 const float* __restrict__ bias, float* __restrict__ Out,
                 int accum, int dorelu) {
  const size_t m0 = (size_t)blockIdx.x * 128;
  __shared__ float As[128][PAD];
  __shared__ float Bt[64][PAD];

  const int tid = threadIdx.x;
  const int wave = tid >> 5, lane = tid & 31;
  const int mrow = lane & 15;
  const int khalf = (lane >> 4) * 2;

#pragma unroll
  for (int rep = 0; rep < 16; ++rep) {         // A: 128x64 = 2048 float4
    int idx = tid + rep * 128;
    int r = idx >> 4, c = (idx & 15) * 4;
    *reinterpret_cast<float4*>(&As[r][c]) =
        *reinterpret_cast<const float4*>(X + m0 * 64 + (size_t)idx * 4);
  }
#pragma unroll
  for (int rep = 0; rep < 8; ++rep) {          // Wt: 64x64 = 1024 float4
    int idx = tid + rep * 128;
    int r = idx >> 4, c = (idx & 15) * 4;
    *reinterpret_cast<float4*>(&Bt[r][c]) =
        *reinterpret_cast<const float4*>(Wt + (size_t)idx * 4);
  }
  __syncthreads();

  v8f acc0[4], acc1[4];
#pragma unroll
  for (int t = 0; t < 4; ++t) { acc0[t] = vzero8(); acc1[t] = vzero8(); }

#pragma unroll
  for (int kk = 0; kk < 64; kk += 4) {
    v2f a0 = *reinterpret_cast<const v2f*>(&As[wave * 32 + mrow][kk + khalf]);
    v2f a1 = *reinterpret_cast<const v2f*>(&As[wave * 32 + 16 + mrow][kk + khalf]);
#pragma unroll
    for (int t = 0; t < 4; ++t) {
      v2f b = *reinterpret_cast<const v2f*>(&Bt[t * 16 + mrow][kk + khalf]);
      acc0[t] = wmma_f32x4(a0, b, acc0[t]);
      acc1[t] = wmma_f32x4(a1, b, acc1[t]);
    }
  }

  const int n = lane & 15;
  const size_t mb0 = m0 + wave * 32 + (lane >> 4) * 8;
#pragma unroll
  for (int t = 0; t < 4; ++t) {
    int col = t * 16 + n;
    float vb = bias ? bias[col] : 0.f;
#pragma unroll
    for (int r = 0; r < 8; ++r) {
      size_t off = (mb0 + r) * 64 + col;
      float v = acc0[t][r] + vb;
      if (accum) v += Out[off];
      if (dorelu) v = fmaxf(v, 0.f);
      Out[off] = v;
      size_t off2 = off + (size_t)16 * 64;
      float v2 = acc1[t][r] + vb;
      if (accum) v2 += Out[off2];
      if (dorelu) v2 = fmaxf(v2, 0.f);
      Out[off2] = v2;
    }
  }
}

// ---------------- temporal GLU conv, cin=64, via WMMA -----------------------
// in: (B, S_in, N, 64) -> out: (B, S_in-2, N, 64).
// wt1/wt2/wt3: PRE-TRANSPOSED weights, (3, 64[f], 64[k]) each.
// Optional fused BatchNorm on the input (per-node affine), applied while
// loading the A tile: x = (x - mean[n]) * inv[n] * gamma[n] + beta[n].
__global__ __launch_bounds__(128, 1)
void k_tc64(const float* __restrict__ in,
            const float* __restrict__ wt1, const float* __restrict__ wt2,
            const float* __restrict__ wt3,
            const float* __restrict__ b1, const float* __restrict__ b2,
            const float* __restrict__ b3,
            const float* __restrict__ bn_mean, const float* __restrict__ bn_inv,
            const float* __restrict__ bn_gamma, const float* __restrict__ bn_beta,
            float* __restrict__ out, int S_in) {
  const int S_out = S_in - 2;
  const int n0 = blockIdx.x * 64;
  const int so = blockIdx.y;
  const int b  = blockIdx.z;

  __shared__ float As[64][PAD];
  __shared__ float Bt[3][64][PAD];

  const int tid = threadIdx.x;
  const int wave = tid >> 5, lane = tid & 31;
  const int mrow = lane & 15;
  const int khalf = (lane >> 4) * 2;
  const bool has_bn = (bn_mean != nullptr);

  v8f acc1[4], acc2[4], acc3[4];
#pragma unroll
  for (int t = 0; t < 4; ++t) { acc1[t] = vzero8(); acc2[t] = vzero8(); acc3[t] = vzero8(); }

  for (int dt = 0; dt < 3; ++dt) {
    const float* src = in + ((size_t)(b * S_in + so + dt) * N_NODES + n0) * HID;
#pragma unroll
    for (int rep = 0; rep < 8; ++rep) {        // A: 64x64 = 1024 float4
      int idx = tid + rep * 128;
      int r = idx >> 4, c = (idx & 15) * 4;
      float4 v = *reinterpret_cast<const float4*>(src + (size_t)idx * 4);
      if (has_bn) {
        int nb = n0 + r;
        float sc = bn_inv[nb] * bn_gamma[nb];
        float sh = fmaf(-bn_mean[nb], sc, bn_beta[nb]);
        v.x = fmaf(v.x, sc, sh); v.y = fmaf(v.y, sc, sh);
        v.z = fmaf(v.z, sc, sh); v.w = fmaf(v.w, sc, sh);
      }
      *reinterpret_cast<float4*>(&As[r][c]) = v;
    }
    const float* wsrc[3] = {wt1 + dt * 4096, wt2 + dt * 4096, wt3 + dt * 4096};
#pragma unroll
    for (int w = 0; w < 3; ++w) {
#pragma unroll
      for (int rep = 0; rep < 8; ++rep) {
        int idx = tid + rep * 128;
        int r = idx >> 4, c = (idx & 15) * 4;
        *reinterpret_cast<float4*>(&Bt[w][r][c]) =
            *reinterpret_cast<const float4*>(wsrc[w] + (size_t)idx * 4);
      }
    }
    __syncthreads();
#pragma unroll
    for (int kk = 0; kk < 64; kk += 4) {
      v2f a = *reinterpret_cast<const v2f*>(&As[wave * 16 + mrow][kk + khalf]);
#pragma unroll
      for (int t = 0; t < 4; ++t) {
        v2f bb;
        bb = *reinterpret_cast<const v2f*>(&Bt[0][t * 16 + mrow][kk + khalf]);
        acc1[t] = wmma_f32x4(a, bb, acc1[t]);
        bb = *reinterpret_cast<const v2f*>(&Bt[1][t * 16 + mrow][kk + khalf]);
        acc2[t] = wmma_f32x4(a, bb, acc2[t]);
        bb = *reinterpret_cast<const v2f*>(&Bt[2][t * 16 + mrow][kk + khalf]);
        acc3[t] = wmma_f32x4(a, bb, acc3[t]);
      }
    }
    __syncthreads();
  }

  const int n = lane & 15;
  const int nbase = n0 + wave * 16 + (lane >> 4) * 8;
  float* dst = out + (size_t)(b * S_out + so) * N_NODES * HID;
#pragma unroll
  for (int t = 0; t < 4; ++t) {
    int col = t * 16 + n;
    float vb1 = b1[col], vb2 = b2[col], vb3 = b3[col];
#pragma unroll
    for (int r = 0; r < 8; ++r) {
      float c1 = acc1[t][r] + vb1;
      float c2 = acc2[t][r] + vb2;
      float c3 = acc3[t][r] + vb3;
      float g = c1 * (1.f / (1.f + __expf(-c2))) + c3;
      dst[(size_t)(nbase + r) * HID + col] = fmaxf(g, 0.f);
    }
  }
}

// ---------------- batch-norm statistics per node ----------------------------
__global__ void k_bn_stats(const float* __restrict__ t, float* __restrict__ mean,
                           float* __restrict__ inv, int BS) {
  __shared__ float r1[256], r2[256];
  const int nn = blockIdx.x;
  const int tid = threadIdx.x;
  float s = 0.f, s2 = 0.f;
  const int cnt = BS * 64;
  for (int i = tid; i < cnt; i += 256) {
    int p = i >> 6, c = i & 63;
    float v = t[((size_t)p * N_NODES + nn) * 64 + c];
    s += v; s2 += v * v;
  }
  r1[tid] = s; r2[tid] = s2;
  __syncthreads();
  for (int off = 128; off > 0; off >>= 1) {
    if (tid < off) { r1[tid] += r1[tid + off]; r2[tid] += r2[tid + off]; }
    __syncthreads();
  }
  if (tid == 0) {
    float m = r1[0] / (float)cnt;
    float var = r2[0] / (float)cnt - m * m;
    mean[nn] = m;
    inv[nn] = rsqrtf(var + 1e-5f);
  }
}

// ---------------- head: slice last time step + linear 64->12 ----------------
__global__ void k_head(const float* __restrict__ t, const float* __restrict__ W,
                       const float* __restrict__ bias, float* __restrict__ out) {
  int i = blockIdx.x * 256 + threadIdx.x;
  const int total = BATCH * N_NODES * 12;
  if (i >= total) return;
  int f = i % 12;
  int n = (i / 12) & 2047;
  int b = i / (12 * N_NODES);
  const float* row = t + (((size_t)b * 14 + 13) * N_NODES + n) * HID;
  float s = bias[f];
#pragma unroll
  for (int h = 0; h < 64; ++h) s = fmaf(row[h], W[h * 12 + f], s);
  out[i] = s;
}

// ---------------------------------------------------------------------------
extern "C" void kernel_launch(void* const* d_in, const int* in_sizes, int n_in,
                              void* d_out, int out_size, void* d_ws,
                              size_t ws_size, hipStream_t stream) {
  (void)in_sizes; (void)n_in; (void)out_size; (void)ws_size;
  // Input order: top-level insertion order {x, adjacency, params}; params
  // flattened jax-style (dict keys sorted at each level, lists in order):
  // blocks[i]: bn.beta, bn.gamma, cheb.W, cheb.b, tc1.{b1,b2,b3,w1,w2,w3},
  //            tc2.{b1,b2,b3,w1,w2,w3}; then final.{b1,b2,b3,w1,w2,w3}; lin.{W,b}
  const float* x   = (const float*)d_in[0];
  const float* adj = (const float*)d_in[1];
  auto in = [&](int i) { return (const float*)d_in[i]; };
  const int B0 = 2, B1 = 18;

  float* ws     = (float*)d_ws;
  float* w_norm = ws;                     // 2048*2048
  float* w_dis  = w_norm + 4194304;       // 2048
  float* w_mean = w_dis + 2048;           // 2048
  float* w_inv  = w_mean + 2048;          // 2048
  float* w_x0   = w_inv + 2048;           // 16*24*2048*2
  float* w_wt   = w_x0 + 1572864;         // 42 * 4096 transposed weights
  float* bufA   = w_wt + 42 * 4096;       // 16*22*2048*64 each
  float* bufB   = bufA + 46137344;
  float* bufC   = bufB + 46137344;
  auto wt = [&](int m) { return w_wt + (size_t)m * 4096; };

  dim3 blk(256);
  // Laplacian + input permute
  k_dis<<<N_NODES, blk, 0, stream>>>(adj, w_dis);
  k_norm<<<(4194304 + 255) / 256, blk, 0, stream>>>(adj, w_dis, w_norm);
  k_permute<<<(1572864 + 255) / 256, blk, 0, stream>>>(x, w_x0);

  // Pre-transpose all 64x64 weight matrices.
  // mats: 0..2 cheb0 | 3..11 blk0.tc2 {w1,w2,w3} | 12..20 blk1.tc1 |
  //       21..23 cheb1 | 24..32 blk1.tc2 | 33..41 final
  k_transpose64<<<3, blk, 0, stream>>>(in(B0 + 2), wt(0));
  k_transpose64<<<3, blk, 0, stream>>>(in(B0 + 13), wt(3));
  k_transpose64<<<3, blk, 0, stream>>>(in(B0 + 14), wt(6));
  k_transpose64<<<3, blk, 0, stream>>>(in(B0 + 15), wt(9));
  k_transpose64<<<3, blk, 0, stream>>>(in(B1 + 7),  wt(12));
  k_transpose64<<<3, blk, 0, stream>>>(in(B1 + 8),  wt(15));
  k_transpose64<<<3, blk, 0, stream>>>(in(B1 + 9),  wt(18));
  k_transpose64<<<3, blk, 0, stream>>>(in(B1 + 2),  wt(21));
  k_transpose64<<<3, blk, 0, stream>>>(in(B1 + 13), wt(24));
  k_transpose64<<<3, blk, 0, stream>>>(in(B1 + 14), wt(27));
  k_transpose64<<<3, blk, 0, stream>>>(in(B1 + 15), wt(30));
  k_transpose64<<<3, blk, 0, stream>>>(in(34 + 3),  wt(33));
  k_transpose64<<<3, blk, 0, stream>>>(in(34 + 4),  wt(36));
  k_transpose64<<<3, blk, 0, stream>>>(in(34 + 5),  wt(39));

  // ----- block 0 (S: 24 -> 22 -> 20) -----
  k_tc_cin2<<<(46137344 + 255) / 256, blk, 0, stream>>>(
      w_x0, in(B0 + 7), in(B0 + 4), in(B0 + 8), in(B0 + 5), in(B0 + 9),
      in(B0 + 6), bufA);  // Tx0 = bufA (S=22)
  {
    const int G = BATCH * 22, Mg = G * N_NODES;
    const float* chb = in(B0 + 3);
    k_chan_gemm<<<Mg / 128, 128, 0, stream>>>(bufA, wt(0), nullptr, bufB, 0, 0);
    k_graph_gemm<<<dim3(16, G), 128, 0, stream>>>(w_norm, bufA, bufA, bufC, 1.f, 0.f);
    k_chan_gemm<<<Mg / 128, 128, 0, stream>>>(bufC, wt(1), nullptr, bufB, 1, 0);
    k_graph_gemm<<<dim3(16, G), 128, 0, stream>>>(w_norm, bufC, bufA, bufA, 2.f, -1.f);
    k_chan_gemm<<<Mg / 128, 128, 0, stream>>>(bufA, wt(2), chb, bufB, 1, 1);
  }
  k_tc64<<<dim3(32, 20, BATCH), 128, 0, stream>>>(
      bufB, wt(3), wt(6), wt(9), in(B0 + 10), in(B0 + 11), in(B0 + 12),
      nullptr, nullptr, nullptr, nullptr, bufC, 22);  // tc2: S 22 -> 20
  k_bn_stats<<<N_NODES, blk, 0, stream>>>(bufC, w_mean, w_inv, BATCH * 20);

  // ----- block 1 (S: 20 -> 18 -> 16); BN0 fused into tc1's loads -----
  k_tc64<<<dim3(32, 18, BATCH), 128, 0, stream>>>(
      bufC, wt(12), wt(15), wt(18), in(B1 + 4), in(B1 + 5), in(B1 + 6),
      w_mean, w_inv, in(B0 + 1), in(B0 + 0), bufA, 20);  // Tx0 = bufA
  {
    const int G = BATCH * 18, Mg = G * N_NODES;
    const float* chb = in(B1 + 3);
    k_chan_gemm<<<Mg / 128, 128, 0, stream>>>(bufA, wt(21), nullptr, bufB, 0, 0);
    k_graph_gemm<<<dim3(16, G), 128, 0, stream>>>(w_norm, bufA, bufA, bufC, 1.f, 0.f);
    k_chan_gemm<<<Mg / 128, 128, 0, stream>>>(bufC, wt(22), nullptr, bufB, 1, 0);
    k_graph_gemm<<<dim3(16, G), 128, 0, stream>>>(w_norm, bufC, bufA, bufA, 2.f, -1.f);
    k_chan_gemm<<<Mg / 128, 128, 0, stream>>>(bufA, wt(23), chb, bufB, 1, 1);
  }
  k_tc64<<<dim3(32, 16, BATCH), 128, 0, stream>>>(
      bufB, wt(24), wt(27), wt(30), in(B1 + 10), in(B1 + 11), in(B1 + 12),
      nullptr, nullptr, nullptr, nullptr, bufC, 18);  // tc2: S 18 -> 16
  k_bn_stats<<<N_NODES, blk, 0, stream>>>(bufC, w_mean, w_inv, BATCH * 16);

  // ----- final tc (S: 16 -> 14, BN1 fused) + head -----
  k_tc64<<<dim3(32, 14, BATCH), 128, 0, stream>>>(
      bufC, wt(33), wt(36), wt(39), in(34), in(35), in(36),
      w_mean, w_inv, in(B1 + 1), in(B1 + 0), bufA, 16);
  k_head<<<(BATCH * N_NODES * 12 + 255) / 256, blk, 0, stream>>>(
      bufA, in(40), in(41), (float*)d_out);
}